// Attention_29437705847166
// MI455X (gfx1250) — compile-verified
//
#include <hip/hip_runtime.h>
#include <hip/hip_bf16.h>

// Problem constants from the reference
constexpr int Lh = 200;   // history length
constexpr int Dd = 128;   // feature dim
constexpr int Hh = 128;   // hidden dim
constexpr int LP = 208;   // L padded to 13 * 16 WMMA tiles
constexpr int VS = 132;   // padded LDS row stride in floats (bank-conflict free)
constexpr float MASK_FILL = -4294967295.0f;  // float(-2**32 + 1)

typedef float v2f __attribute__((ext_vector_type(2)));
typedef float v8f __attribute__((ext_vector_type(8)));
typedef int   v4i __attribute__((ext_vector_type(4)));

// Pointer types the async-to-LDS builtin expects (from hipcc's diagnostic:
// param 0 is 'int __vector(4) __device__ *', i.e. AS(1) int4*; param 1 is the LDS side, AS(3)).
typedef __attribute__((address_space(1))) v4i* async_gptr_t;
typedef __attribute__((address_space(3))) v4i* async_lptr_t;

__global__ __launch_bounds__(256) void att_fused_kernel(
    const float* __restrict__ Qg, const float* __restrict__ Vg,
    const unsigned char* __restrict__ maskg,
    const float* __restrict__ W1g, const float* __restrict__ b1g,
    const float* __restrict__ W2g, const float* __restrict__ b2g,
    float* __restrict__ outg)
{
  __shared__ float Vs[LP * VS];     // V[b] tile, padded stride
  __shared__ float Qs[Dd];          // Q[b]
  __shared__ float qc_s[Hh];        // b1 + Q[b] @ W1_top  (broadcast over L)
  __shared__ float score_s[LP];     // per-row pre-bias scores
  __shared__ float alpha_s[Lh];     // softmax weights
  __shared__ float red_s[256];      // reduction scratch

  const int tid  = threadIdx.x;
  const int lane = tid & 31;
  const int wid  = tid >> 5;        // 8 waves, wave w owns H columns [16w, 16w+16)
  const int b    = blockIdx.x;

  const float* __restrict__ Vrow = Vg + (size_t)b * (Lh * Dd);

  // ---- Stage V[b] (100 KB) into LDS: the single HBM pass over V ----
#if defined(__gfx1250__) && \
    __has_builtin(__builtin_amdgcn_global_load_async_to_lds_b128) && \
    __has_builtin(__builtin_amdgcn_s_wait_asynccnt)
  for (int i = tid; i < (Lh * Dd) / 4; i += 256) {
    const int l = i >> 5;           // (i*4)/128
    const int d = (i & 31) << 2;    // (i*4)%128
    __builtin_amdgcn_global_load_async_to_lds_b128(
        (async_gptr_t)(float*)(Vrow + i * 4),
        (async_lptr_t)(&Vs[l * VS + d]),
        0, 0);
  }
#define ATT_ASYNC_V 1
#else
  for (int i = tid; i < (Lh * Dd) / 4; i += 256) {
    const int l = i >> 5;
    const int d = (i & 31) << 2;
    const float4 v = reinterpret_cast<const float4*>(Vrow)[i];
    *reinterpret_cast<float4*>(&Vs[l * VS + d]) = v;
  }
#endif

  // Zero the 8 pad rows (200..207) so WMMA on them yields finite garbage we discard.
  for (int i = tid; i < (LP - Lh) * Dd; i += 256)
    Vs[(Lh + i / Dd) * VS + (i % Dd)] = 0.f;
  if (tid < Dd) Qs[tid] = Qg[(size_t)b * Dd + tid];
  if (tid < LP) score_s[tid] = 0.f;

#if defined(ATT_ASYNC_V)
  __builtin_amdgcn_s_wait_asynccnt(0);
#endif
  __syncthreads();

  // ---- Q-contribution: qc[h] = b1[h] + sum_d Q[b,d] * W1[d,h]  (tiny, once per batch) ----
  if (tid < Hh) {
    float acc = b1g[tid];
#pragma unroll 8
    for (int d = 0; d < Dd; ++d) acc += Qs[d] * W1g[d * Hh + tid];
    qc_s[tid] = acc;
  }
  __syncthreads();

  // ---- Main GEMM: fc1_v = V[b] @ W1_bot via V_WMMA_F32_16X16X4_F32, fused score reduce ----
  const int n0   = wid * 16;
  const int half = lane >> 4;       // K sub-block selector (A/B VGPR layouts)
  const int mrow = lane & 15;

  // Preload B operand (W1 bottom, this wave's 16 H-columns) into 64 VGPRs, reused 13x.
  v2f Breg[32];
#pragma unroll
  for (int kk = 0; kk < 32; ++kk) {
    const int r = Dd + 4 * kk + 2 * half;            // K rows {2*half, 2*half+1} of 4-block
    Breg[kk][0] = W1g[(r + 0) * Hh + n0 + mrow];
    Breg[kk][1] = W1g[(r + 1) * Hh + n0 + mrow];
  }
  const float qcv = qc_s[n0 + mrow];
  const float w2v = W2g[n0 + mrow];

  for (int t = 0; t < 13; ++t) {
    v8f a0 = {0.f,0.f,0.f,0.f,0.f,0.f,0.f,0.f};
    v8f a1 = a0, a2 = a0, a3 = a0;
    const float* __restrict__ arow = &Vs[(t * 16 + mrow) * VS + 2 * half];
#pragma unroll
    for (int kk = 0; kk < 8; ++kk) {   // 4-way interleaved K for WMMA ILP
      const v2f A0 = *reinterpret_cast<const v2f*>(arow + 4 * kk);
      const v2f A1 = *reinterpret_cast<const v2f*>(arow + 4 * (kk + 8));
      const v2f A2 = *reinterpret_cast<const v2f*>(arow + 4 * (kk + 16));
      const v2f A3 = *reinterpret_cast<const v2f*>(arow + 4 * (kk + 24));
      a0 = __builtin_amdgcn_wmma_f32_16x16x4_f32(false, A0, false, Breg[kk],      (short)0, a0, false, false);
      a1 = __builtin_amdgcn_wmma_f32_16x16x4_f32(false, A1, false, Breg[kk + 8],  (short)0, a1, false, false);
      a2 = __builtin_amdgcn_wmma_f32_16x16x4_f32(false, A2, false, Breg[kk + 16], (short)0, a2, false, false);
      a3 = __builtin_amdgcn_wmma_f32_16x16x4_f32(false, A3, false, Breg[kk + 24], (short)0, a3, false, false);
    }
    const v8f acc = a0 + a1 + a2 + a3;

    // Fused score: score[l] += sum_{n in wave's 16 cols} relu(fc1)*W2[n]
    // C layout: VGPR i -> row t*16 + i (lanes 0-15) / t*16 + i + 8 (lanes 16-31), col n0+mrow.
#pragma unroll
    for (int i = 0; i < 8; ++i) {
      float v = fmaxf(acc[i] + qcv, 0.f) * w2v;
      v += __shfl_xor(v, 1, 16);
      v += __shfl_xor(v, 2, 16);
      v += __shfl_xor(v, 4, 16);
      v += __shfl_xor(v, 8, 16);
      if (mrow == 0) atomicAdd(&score_s[t * 16 + i + half * 8], v);
    }
  }
  __syncthreads();

  // ---- Softmax over L with mask (pad rows 200..207 excluded) ----
  const unsigned char* __restrict__ mrowp = maskg + (size_t)b * Lh;
  const float b2v = b2g[0];
  const bool inr = tid < Lh;
  const unsigned char mk = inr ? mrowp[tid] : (unsigned char)1;
  const float sval = (inr && !mk) ? (score_s[tid] + b2v) : MASK_FILL;

  red_s[tid] = sval;
  __syncthreads();
#pragma unroll
  for (int s = 128; s > 0; s >>= 1) {
    if (tid < s) red_s[tid] = fmaxf(red_s[tid], red_s[tid + s]);
    __syncthreads();
  }
  const float mx = red_s[0];
  __syncthreads();
  const float e = inr ? __expf(sval - mx) : 0.f;   // masked -> exp(-4.29e9) == 0
  red_s[tid] = e;
  __syncthreads();
#pragma unroll
  for (int s = 128; s > 0; s >>= 1) {
    if (tid < s) red_s[tid] += red_s[tid + s];
    __syncthreads();
  }
  const float inv = 1.f / red_s[0];
  __syncthreads();
  if (inr) alpha_s[tid] = e * inv;
  __syncthreads();

  // ---- att[b,:] = sum_l alpha[l] * V[b,l,:]  (V read from LDS, second HBM pass avoided) ----
  const int d   = tid & 127;
  const int hlf = tid >> 7;
  const int l0  = hlf * 100;
  float p = 0.f;
#pragma unroll 4
  for (int l = l0; l < l0 + 100; ++l) p += alpha_s[l] * Vs[l * VS + d];
  red_s[tid] = p;
  __syncthreads();
  if (tid < Dd) outg[(size_t)b * Dd + tid] = red_s[tid] + red_s[tid + 128];
}

extern "C" void kernel_launch(void* const* d_in, const int* in_sizes, int n_in,
                              void* d_out, int out_size, void* d_ws, size_t ws_size,
                              hipStream_t stream) {
  const float*         Q    = (const float*)d_in[0];
  const float*         V    = (const float*)d_in[1];
  const unsigned char* mask = (const unsigned char*)d_in[2];  // jax bool -> 1 byte
  const float*         W1   = (const float*)d_in[3];
  const float*         b1   = (const float*)d_in[4];
  const float*         W2   = (const float*)d_in[5];
  const float*         b2   = (const float*)d_in[6];
  float*               out  = (float*)d_out;

  const int Bn = in_sizes[0] / Dd;  // 2048
  att_fused_kernel<<<dim3(Bn), dim3(256), 0, stream>>>(Q, V, mask, W1, b1, W2, b2, out);
}